// GCN_46162308498193
// MI455X (gfx1250) — compile-verified
//
#include <hip/hip_runtime.h>
#include <math.h>

#define NND   100000          // N_NODES
#define NED   1600000         // N_EDGES
#define DDIM  128
#define NGR   16

typedef __attribute__((ext_vector_type(2))) float v2f;
typedef __attribute__((ext_vector_type(8))) float v8f;

// ---------------------------------------------------------------- zero fill
__global__ void k_zero(float* __restrict__ p, int n) {
  int i = blockIdx.x * blockDim.x + threadIdx.x;
  if (i < n) p[i] = 0.0f;
}

// ------------------------------------------------- projection: (A @ W)*norm
// One wave computes a 16-row x 128-col tile with V_WMMA_F32_16X16X4_F32.
// A rows come either via gather (word_ids -> embedding table) or directly.
__global__ void k_gemm_norm(const float* __restrict__ Asrc,
                            const int*   __restrict__ gatherIdx,
                            const float* __restrict__ W,      // [128][128] row-major (k,n)
                            const float* __restrict__ norm,
                            float*       __restrict__ out,    // [N][128]
                            int numTiles) {
  const int wave = blockIdx.x * (blockDim.x >> 5) + (threadIdx.x >> 5);
  if (wave >= numTiles) return;               // wave-uniform: EXEC stays all-1s
  const int lane = threadIdx.x & 31;
  const int half = lane >> 4;                 // 0 or 1
  const int lm   = lane & 15;                 // row (A) / col (B) within tile
  const int rowBase = wave * 16;

  const int m = rowBase + lm;
  const float* arow = gatherIdx ? (Asrc + (size_t)gatherIdx[m] * DDIM)
                                : (Asrc + (size_t)m * DDIM);

  v8f acc[8] = {};
  for (int k = 0; k < DDIM; k += 4) {
    // A 16x4 f32 fragment: VGPR0 = K(0|2), VGPR1 = K(1|3) per lane-half
    v2f a;
    a.x = arow[k + 2 * half];
    a.y = arow[k + 2 * half + 1];
#pragma unroll
    for (int t = 0; t < 8; ++t) {
      // B 4x16 f32 fragment: row striped across lanes, lane-half selects K+2
      v2f b;
      b.x = W[(size_t)(k + 2 * half)     * DDIM + t * 16 + lm];
      b.y = W[(size_t)(k + 2 * half + 1) * DDIM + t * 16 + lm];
      acc[t] = __builtin_amdgcn_wmma_f32_16x16x4_f32(
          false, a, false, b, (short)0, acc[t], false, false);
    }
  }

  // epilogue: scale by norm[row], store. C/D layout: vgpr i holds row i+8*half.
  float nv[8];
#pragma unroll
  for (int i = 0; i < 8; ++i) nv[i] = norm[rowBase + i + 8 * half];
#pragma unroll
  for (int t = 0; t < 8; ++t) {
#pragma unroll
    for (int i = 0; i < 8; ++i) {
      out[(size_t)(rowBase + i + 8 * half) * DDIM + t * 16 + lm] = acc[t][i] * nv[i];
    }
  }
}

// --------------------------------------------- edge scatter: agg[dst] += t[src]*w
// One wave per edge; lane handles 4 consecutive features (float4 gather,
// 4x global_atomic_add_f32 — L2-resident working set on MI455X).
__global__ void k_scatter(const float* __restrict__ t,
                          const int*   __restrict__ src,
                          const int*   __restrict__ dst,
                          const float* __restrict__ ew,
                          float*       __restrict__ agg) {
  const int e = blockIdx.x * (blockDim.x >> 5) + (threadIdx.x >> 5);
  if (e >= NED) return;
  const int lane = threadIdx.x & 31;
  const float w = ew[e];
  const float4 v = *(const float4*)(t + (size_t)src[e] * DDIM + lane * 4);
  float* o = agg + (size_t)dst[e] * DDIM + lane * 4;
  atomicAdd(o + 0, v.x * w);
  atomicAdd(o + 1, v.y * w);
  atomicAdd(o + 2, v.z * w);
  atomicAdd(o + 3, v.w * w);
}

// -------------------------------------------- h = relu(h*norm + b), in place
__global__ void k_relu_bias(float* __restrict__ h,
                            const float* __restrict__ norm,
                            const float* __restrict__ b) {
  int i = blockIdx.x * blockDim.x + threadIdx.x;
  if (i >= NND * DDIM) return;
  int node = i >> 7, col = i & 127;
  h[i] = fmaxf(h[i] * norm[node] + b[col], 0.0f);
}

// ------------------------------------- per-graph max pool (values >= 0 post-relu)
__global__ void k_pool(const float* __restrict__ h,
                       const int*   __restrict__ gid,
                       unsigned*    __restrict__ g) {
  int i = blockIdx.x * blockDim.x + threadIdx.x;
  if (i >= NND * DDIM) return;
  int node = i >> 7, col = i & 127;
  atomicMax(g + (size_t)gid[node] * DDIM + col, __float_as_uint(h[i]));
}

// -------------------- head: z = g@Wout + bout, BCE-with-logits loss, sigmoid
__global__ void k_final(const float* __restrict__ g,
                        const float* __restrict__ Wout,
                        const float* __restrict__ bout,
                        const float* __restrict__ y,
                        float* __restrict__ out) {  // out[0]=loss, out[1..16]=y_pred
  __shared__ float sl[NGR];
  const int w = threadIdx.x >> 5;     // one wave32 per graph (16 waves)
  const int lane = threadIdx.x & 31;
  float s = 0.0f;
  for (int j = lane; j < DDIM; j += 32) s += g[w * DDIM + j] * Wout[j];
  for (int off = 16; off > 0; off >>= 1) s += __shfl_down(s, off, 32);
  if (lane == 0) {
    float z = s + bout[0];
    out[1 + w] = 1.0f / (1.0f + expf(-z));
    sl[w] = fmaxf(z, 0.0f) - z * y[w] + log1pf(expf(-fabsf(z)));
  }
  __syncthreads();
  if (threadIdx.x == 0) {
    float t = 0.0f;
#pragma unroll
    for (int i = 0; i < NGR; ++i) t += sl[i];
    out[0] = t / (float)NGR;
  }
}

// ---------------------------------------------------------------------------
extern "C" void kernel_launch(void* const* d_in, const int* in_sizes, int n_in,
                              void* d_out, int out_size, void* d_ws, size_t ws_size,
                              hipStream_t stream) {
  const int*   word_ids    = (const int*)  d_in[0];
  const int*   edge_src    = (const int*)  d_in[1];
  const int*   edge_dst    = (const int*)  d_in[2];
  const float* edge_weight = (const float*)d_in[3];
  const float* norm        = (const float*)d_in[4];
  const int*   graph_id    = (const int*)  d_in[5];
  const float* y_data      = (const float*)d_in[6];
  const float* word_embeds = (const float*)d_in[7];
  const float* W0          = (const float*)d_in[8];
  const float* b0          = (const float*)d_in[9];
  const float* W1          = (const float*)d_in[10];
  const float* b1          = (const float*)d_in[11];
  const float* Wout        = (const float*)d_in[12];
  const float* bout        = (const float*)d_in[13];

  float* bufA = (float*)d_ws;                        // projection output t
  float* bufB = bufA + (size_t)NND * DDIM;           // aggregation / h
  float* gbuf = bufB + (size_t)NND * DDIM;           // [16][128] pooled

  const int TILES   = NND / 16;                      // 6250 exact
  const int gemmBlk = (TILES + 7) / 8;               // 8 waves / 256-thr block
  const int edgeBlk = (NED + 7) / 8;                 // 8 edges / 256-thr block
  const int elemBlk = (NND * DDIM + 255) / 256;

  // ---- layer 0 ----
  k_zero<<<elemBlk, 256, 0, stream>>>(bufB, NND * DDIM);
  k_gemm_norm<<<gemmBlk, 256, 0, stream>>>(word_embeds, word_ids, W0, norm, bufA, TILES);
  k_scatter<<<edgeBlk, 256, 0, stream>>>(bufA, edge_src, edge_dst, edge_weight, bufB);
  k_relu_bias<<<elemBlk, 256, 0, stream>>>(bufB, norm, b0);

  // ---- layer 1 ----
  k_gemm_norm<<<gemmBlk, 256, 0, stream>>>(bufB, nullptr, W1, norm, bufA, TILES);
  // zero bufB and gbuf (contiguous) before re-accumulating / pooling
  k_zero<<<(NND * DDIM + NGR * DDIM + 255) / 256, 256, 0, stream>>>(bufB, NND * DDIM + NGR * DDIM);
  k_scatter<<<edgeBlk, 256, 0, stream>>>(bufA, edge_src, edge_dst, edge_weight, bufB);
  k_relu_bias<<<elemBlk, 256, 0, stream>>>(bufB, norm, b1);

  // ---- pooling + head ----
  k_pool<<<elemBlk, 256, 0, stream>>>(bufB, graph_id, (unsigned*)gbuf);
  k_final<<<1, 512, 0, stream>>>(gbuf, Wout, bout, y_data, (float*)d_out);
}